// QuantumAttentionHead_65481071402349
// MI455X (gfx1250) — compile-verified
//
#include <hip/hip_runtime.h>
#include <cstdint>

// QuantumAttentionHead — closed form via Clifford back-propagation.
//
// The reference circuit = RX layer + fixed Clifford (CNOT chains + H, depth 2)
// + <Z_q>.  Conjugating each Z_q backward through the Clifford yields pure
// Z-strings (verified against basis cases), so with c_j = cos(x_j):
//   out = [c0, c1, c0c2, c1c3, c0c2c4, c1c3c5, c0c2c4c6, c0c1c2c3c4c5c6c7]
// Kernel is bandwidth-bound: 16 MB total @ 23.3 TB/s (~0.7 us floor).
// Input is streamed HBM -> LDS with CDNA5 async-copy instructions
// (global_load_async_to_lds_b128, ASYNCcnt / s_wait_asynccnt), read back with
// ds_load_b128 via explicit addrspace(3) pointers, results stored with
// nontemporal b128 stores.  No matmul structure survives -> WMMA N/A.

typedef float v4f __attribute__((ext_vector_type(4)));
typedef __attribute__((address_space(3))) v4f* lds_v4f_ptr;

#define THREADS 256

__global__ __launch_bounds__(THREADS) void qhead_closed_form(
    const float* __restrict__ x, float* __restrict__ out, int B) {
  // One row (8 floats, 32 B) per thread; 8 KB tile per block (2 v4f per row).
  __shared__ __align__(16) v4f tile[THREADS * 2];

  const int lane      = threadIdx.x & 31;
  const int wave      = threadIdx.x >> 5;
  const int wave_row0 = blockIdx.x * THREADS + wave * 32;
  const int row       = wave_row0 + lane;

  // addrspace(3) view of the tile: ptrtoint gives the raw LDS byte offset
  // (no src_shared_base / generic-pointer detour), and loads through it
  // lower to ds_load_b128 instead of flat loads.
  lds_v4f_ptr lt = (lds_v4f_ptr)tile;

  v4f a, b;
  const bool wave_full = (wave_row0 + 32) <= B;
  if (wave_full) {
    // CDNA5 async bulk copy: each wave streams its own 1 KB chunk of x
    // directly HBM -> LDS, 16 B per lane per instruction (512 B coalesced
    // per wave per op), tracked by ASYNCcnt — no VGPR staging.
    const uint64_t ga0 = (uint64_t)(uintptr_t)x + (uint64_t)wave_row0 * 32u
                       + (uint32_t)(lane * 16);
    const uint64_t ga1 = ga0 + 512u;
    const uint32_t lw  = (uint32_t)(uintptr_t)(lt + wave * 64);  // LDS byte offset
    const uint32_t la0 = lw + (uint32_t)(lane * 16);
    const uint32_t la1 = la0 + 512u;
    asm volatile("global_load_async_to_lds_b128 %0, %1, off"
                 :: "v"(la0), "v"(ga0) : "memory");
    asm volatile("global_load_async_to_lds_b128 %0, %1, off"
                 :: "v"(la1), "v"(ga1) : "memory");
    asm volatile("s_wait_asynccnt 0" ::: "memory");

    // Same-wave producer/consumer: after s_wait_asynccnt 0 the LDS data is
    // visible; fetch this thread's row (2x ds_load_b128).
    const int t2 = (wave * 32 + lane) * 2;
    a = lt[t2 + 0];
    b = lt[t2 + 1];
  } else {
    // Tail wave (only if B % 32 != 0): plain guarded global loads.
    if (row >= B) return;
    const v4f* gp = (const v4f*)(x + (size_t)row * 8);
    a = gp[0];
    b = gp[1];
  }

  // 8 native cosines (v_cos_f32); error ~1e-6, below the reference's own
  // accumulated fp32 statevector error.
  const float c0 = __cosf(a.x);
  const float c1 = __cosf(a.y);
  const float c2 = __cosf(a.z);
  const float c3 = __cosf(a.w);
  const float c4 = __cosf(b.x);
  const float c5 = __cosf(b.y);
  const float c6 = __cosf(b.z);
  const float c7 = __cosf(b.w);

  const float e2 = c0 * c2;        // <Z2>
  const float o3 = c1 * c3;        // <Z3>
  const float e4 = e2 * c4;        // <Z4>
  const float o5 = o3 * c5;        // <Z5>
  const float e6 = e4 * c6;        // <Z6>
  const float z7 = e6 * (o5 * c7); // <Z7> = product of all 8 cosines

  v4f r0 = {c0, c1, e2, o3};
  v4f r1 = {e4, o5, e6, z7};

  // Streaming output, never re-read: nontemporal b128 stores.
  v4f* op = (v4f*)(out + (size_t)row * 8);
  __builtin_nontemporal_store(r0, op + 0);
  __builtin_nontemporal_store(r1, op + 1);
}

extern "C" void kernel_launch(void* const* d_in, const int* in_sizes, int n_in,
                              void* d_out, int out_size, void* d_ws, size_t ws_size,
                              hipStream_t stream) {
  const float* x = (const float*)d_in[0];
  float* out = (float*)d_out;
  const int B = in_sizes[0] / 8;  // 262144 rows of 8 angles
  const int blocks = (B + THREADS - 1) / THREADS;
  qhead_closed_form<<<blocks, THREADS, 0, stream>>>(x, out, B);
}